// SwitchTransformerRouting_24713241821313
// MI455X (gfx1250) — compile-verified
//
#include <hip/hip_runtime.h>
#include <hip/hip_bf16.h>

// Switch-Transformer top-1 routing for MI455X (gfx1250, wave32).
// B=8 S=1024 H=2048 E=16 -> N=8192, C=640.
static constexpr int Hdim = 2048;
static constexpr int Edim = 16;
static constexpr int Ntok = 8192;
static constexpr int CAP  = 640;                 // int(1.25 * N / E)
static constexpr long NEC = (long)Ntok * Edim * CAP;   // 83,886,080
static constexpr int NBLK = 64;                  // route/rank grid

typedef float v2f __attribute__((ext_vector_type(2)));
typedef float v8f __attribute__((ext_vector_type(8)));

// ---------------------------------------------------------------- init
__global__ void smoe_init_kernel(int* counts) {
    if (threadIdx.x < Edim) counts[threadIdx.x] = 0;
}

// ---------------------------------------------------------------- route (WMMA)
// 64 blocks x 256 threads (8 waves). Wave w of block b computes the
// 16-token x 16-expert logits tile for tokens (b*8+w)*16 .. +15 with
// V_WMMA_F32_16X16X4_F32, then 128 threads/block do softmax/argmax.
__global__ __launch_bounds__(256) void smoe_route_kernel(
    const float* __restrict__ hs, const float* __restrict__ rw,
    const float* __restrict__ noise,
    float* __restrict__ gate, int* __restrict__ eidx,
    int* __restrict__ counts, float* __restrict__ cs_part,
    float* __restrict__ z_part)
{
    __shared__ float lds_logits[128][Edim];   // later overwritten with probs
    __shared__ float lds_z[128];

    const int tid  = threadIdx.x;
    const int wave = tid >> 5;
    const int lane = tid & 31;
    const int m    = lane & 15;                // token row (A) / expert col (B)
    const bool hi  = lane >= 16;               // K-offset selector
    const int tok0 = (blockIdx.x * 8 + wave) * 16;

    // Per-lane base pointers (8B-aligned: rows are 8KB-aligned, ka is even).
    const float2* x2 = reinterpret_cast<const float2*>(hs    + (long)(tok0 + m) * Hdim);
    const float2* n2 = reinterpret_cast<const float2*>(noise + (long)(tok0 + m) * Hdim);
    const float2* w2 = reinterpret_cast<const float2*>(rw    + (long)m * Hdim);
    const int koff = hi ? 1 : 0;               // float2 index offset (= 2 floats)

    v8f acc = {0.f, 0.f, 0.f, 0.f, 0.f, 0.f, 0.f, 0.f};

#pragma unroll 4
    for (int k2 = 0; k2 < Hdim / 2; k2 += 2) { // K advances by 4 floats / iter
        const float2 xv = x2[k2 + koff];
        const float2 nv = n2[k2 + koff];
        const float2 wv = w2[k2 + koff];
        v2f a, b;
        a.x = xv.x + (nv.x * 2.0f - 1.0f) * 0.1f;   // jitter in [-eps, eps]
        a.y = xv.y + (nv.y * 2.0f - 1.0f) * 0.1f;
        b.x = wv.x;
        b.y = wv.y;
        acc = __builtin_amdgcn_wmma_f32_16x16x4_f32(
            /*neg_a=*/false, a, /*neg_b=*/false, b,
            /*c_mod=*/(short)0, acc, /*reuse_a=*/false, /*reuse_b=*/false);
    }

    // Spill the 16x16 logits tile: lane L -> expert L%16, VGPR v -> token v(+8).
#pragma unroll
    for (int v = 0; v < 8; ++v) {
        const int tk = wave * 16 + (hi ? v + 8 : v);
        lds_logits[tk][m] = acc[v];
    }
    __syncthreads();

    // 128 tokens per block: softmax over 16 experts, top-1 gate/argmax, z partial.
    if (tid < 128) {
        const int i = blockIdx.x * 128 + tid;
        float lg[Edim];
        float mx = -3.4e38f;
        float zs = 0.f;
#pragma unroll
        for (int e = 0; e < Edim; ++e) {
            lg[e] = lds_logits[tid][e];
            zs += lg[e] * lg[e];
            mx = fmaxf(mx, lg[e]);
        }
        float sum = 0.f;
        float pr[Edim];
#pragma unroll
        for (int e = 0; e < Edim; ++e) { pr[e] = __expf(lg[e] - mx); sum += pr[e]; }
        const float inv = 1.0f / sum;
        int ei = 0;
#pragma unroll
        for (int e = Edim - 1; e >= 0; --e)     // first max index (argmax tie rule)
            if (lg[e] == mx) ei = e;
#pragma unroll
        for (int e = 0; e < Edim; ++e) lds_logits[tid][e] = pr[e] * inv;
        gate[i] = inv;                          // max prob = exp(0)/sum
        eidx[i] = ei;
        atomicAdd(&counts[ei], 1);
        lds_z[tid] = zs;
    }
    __syncthreads();

    // Deterministic per-block partials for the losses.
    if (tid < Edim) {
        float s = 0.f;
        for (int t = 0; t < 128; ++t) s += lds_logits[t][tid];
        cs_part[blockIdx.x * Edim + tid] = s;
    }
    if (tid == 0) {
        float s = 0.f;
        for (int t = 0; t < 128; ++t) s += lds_z[t];
        z_part[blockIdx.x] = s;
    }
}

// ---------------------------------------------------------------- loss reduce + tail
__global__ void smoe_reduce_kernel(const float* __restrict__ cs_part,
                                   const float* __restrict__ z_part,
                                   const int* __restrict__ counts,
                                   float* __restrict__ out_tail)
{
    __shared__ float cs[Edim];
    const int t = threadIdx.x;
    if (t < Edim) {
        float s = 0.f;
        for (int b = 0; b < NBLK; ++b) s += cs_part[b * Edim + t];
        cs[t] = s;
        const int c = counts[t];
        out_tail[t] = (float)(c < CAP ? c : CAP);          // expert usage
    }
    __syncthreads();
    if (t == 0) {
        float ll = 0.f;
        for (int e = 0; e < Edim; ++e)
            ll += (cs[e] / (float)Ntok) * ((float)counts[e] / (float)Ntok);
        out_tail[Edim] = (float)Edim * ll;                 // load balance loss
        float zs = 0.f;
        for (int b = 0; b < NBLK; ++b) zs += z_part[b];
        out_tail[Edim + 1] = zs / (float)((long)Ntok * Edim); // z loss
    }
}

// ---------------------------------------------------------------- rank within expert
// pos[i] = #{ j : e_j==e_i && (g_j > g_i || (g_j==g_i && j<i)) }
// == stable lexsort((-gate, eidx)) slot within the expert group.
__global__ __launch_bounds__(128) void smoe_rank_kernel(
    const float* __restrict__ gate, const int* __restrict__ eidx,
    int* __restrict__ pos)
{
    __shared__ float g[Ntok];           // 32 KB
    __shared__ unsigned char e[Ntok];   //  8 KB  (E=16 fits a byte)
    for (int j = threadIdx.x; j < Ntok; j += 128) {
        g[j] = gate[j];
        e[j] = (unsigned char)eidx[j];
    }
    __syncthreads();

    const int i = blockIdx.x * 128 + threadIdx.x;
    const float gi = g[i];
    const unsigned char ei = e[i];
    int r = 0;
#pragma unroll 8
    for (int j = 0; j < Ntok; ++j) {    // uniform j -> broadcast LDS loads
        const float gj = g[j];
        if (e[j] == ei && (gj > gi || (gj == gi && j < i))) ++r;
    }
    pos[i] = r;
}

// ---------------------------------------------------------------- bulk zero (b128)
__global__ void smoe_zero_kernel(float4* __restrict__ out, long n4) {
    const long stride = (long)gridDim.x * blockDim.x;
    const float4 z = {0.f, 0.f, 0.f, 0.f};
    for (long i = (long)blockIdx.x * blockDim.x + threadIdx.x; i < n4; i += stride)
        out[i] = z;
}

// ---------------------------------------------------------------- scatter nonzeros
__global__ void smoe_scatter_kernel(const float* __restrict__ gate,
                                    const int* __restrict__ eidx,
                                    const int* __restrict__ pos,
                                    float* __restrict__ out)
{
    const int i = blockIdx.x * blockDim.x + threadIdx.x;  // token id
    const int p = pos[i];
    if (p < CAP) {
        const long off = (long)i * Edim * CAP + (long)eidx[i] * CAP + p;
        out[off]       = 1.0f;        // dispatch
        out[NEC + off] = gate[i];     // combine
    }
}

// ---------------------------------------------------------------- launcher
extern "C" void kernel_launch(void* const* d_in, const int* in_sizes, int n_in,
                              void* d_out, int out_size, void* d_ws, size_t ws_size,
                              hipStream_t stream) {
    (void)in_sizes; (void)n_in; (void)out_size; (void)ws_size;
    const float* hs    = (const float*)d_in[0];   // [8,1024,2048]
    const float* rw    = (const float*)d_in[1];   // [16,2048]
    const float* noise = (const float*)d_in[2];   // [8192,2048]
    float* out = (float*)d_out;                   // dispatch | combine | usage | ll | zl

    // Workspace layout (floats/ints, 4B each): ~101 KB total.
    float* ws      = (float*)d_ws;
    float* w_gate  = ws;                               // N
    int*   w_eidx  = (int*)(ws + Ntok);                // N
    int*   w_pos   = (int*)(ws + 2 * Ntok);            // N
    int*   w_cnt   = (int*)(ws + 3 * Ntok);            // 16
    float* w_cs    = ws + 3 * Ntok + Edim;             // 64*16
    float* w_z     = ws + 3 * Ntok + Edim + NBLK * Edim; // 64

    smoe_init_kernel<<<1, 32, 0, stream>>>(w_cnt);
    smoe_route_kernel<<<NBLK, 256, 0, stream>>>(hs, rw, noise, w_gate, w_eidx,
                                                w_cnt, w_cs, w_z);
    smoe_reduce_kernel<<<1, 32, 0, stream>>>(w_cs, w_z, w_cnt, out + 2 * NEC);
    smoe_rank_kernel<<<NBLK, 128, 0, stream>>>(w_gate, w_eidx, w_pos);
    smoe_zero_kernel<<<8192, 256, 0, stream>>>((float4*)out, (2 * NEC) / 4);
    smoe_scatter_kernel<<<Ntok / 256, 256, 0, stream>>>(w_gate, w_eidx, w_pos, out);
}